// TransformerScheduler_81106162417768
// MI455X (gfx1250) — compile-verified
//
#include <hip/hip_runtime.h>
#include <math.h>

// ---- problem constants (match reference) ----
#define BATCH      256
#define MAX_OPES   4096
#define NUM_MAS    64
#define F_OPE      8
#define F_MA       6
#define EPSF       1e-5f

#define OPES_ELEMS   (BATCH * MAX_OPES * F_OPE)          // 8388608
#define MAS_ELEMS    (BATCH * NUM_MAS * F_MA)            // 98304
#define PT_PER_B     (MAX_OPES * NUM_MAS)                // 262144
#define OUT_OFF_MAS  OPES_ELEMS                          // 8388608
#define OUT_OFF_PT   (OPES_ELEMS + MAS_ELEMS)            // 8486912

// ---- workspace layout (floats) ----
// [0,        2048)   : opes per-(b,f) sum
// [2048,     4096)   : opes per-(b,f) sumsq
// [4096,     102400) : pt wave partials: (b*128 + wave)*3 + {sum,sumsq,cnt}
// [102400,   103168) : pt final per-b {sum,sumsq,cnt}
#define WS_OPES_SUM  0
#define WS_OPES_SQ   2048
#define WS_PT_PART   4096
#define WS_PT_FINAL  102400

#define PT_BLOCKS_PER_B  16
#define PT_WAVES_PER_BLK 8
#define PT_WAVES_PER_B   (PT_BLOCKS_PER_B * PT_WAVES_PER_BLK)   // 128
#define PT_CHUNKS_PER_WAVE 32                                   // 128*32*64 = 262144

typedef __attribute__((ext_vector_type(2))) float v2f;
typedef __attribute__((ext_vector_type(8))) float v8f;

// ---------------------------------------------------------------------------
// Phase 1: masked per-feature sum/sumsq of raw_opes. One block (256 thr) / batch.
// thread t handles feature f=t%8, rows r = t/8 + 32k  -> fully coalesced reads.
// ---------------------------------------------------------------------------
__global__ void opes_stats_kernel(const float* __restrict__ opes,
                                  const int* __restrict__ nums,
                                  float* __restrict__ ws) {
    const int b = blockIdx.x;
    const int t = threadIdx.x;        // 0..255
    const int f = t & 7;
    const int j = t >> 3;             // 0..31
    const int nv = nums[b];
    const float* base = opes + (size_t)b * (MAX_OPES * F_OPE);

    float s = 0.f, q = 0.f;
    for (int r = j; r < nv; r += 32) {
        float x = base[(size_t)r * F_OPE + f];
        s += x;
        q += x * x;
    }
    __shared__ float ssum[256];
    __shared__ float ssq[256];
    ssum[t] = s; ssq[t] = q;
    __syncthreads();
    for (int off = 128; off >= 8; off >>= 1) {
        if (t < off) { ssum[t] += ssum[t + off]; ssq[t] += ssq[t + off]; }
        __syncthreads();
    }
    if (t < 8) {
        ws[WS_OPES_SUM + b * 8 + t] = ssum[t];
        ws[WS_OPES_SQ  + b * 8 + t] = ssq[t];
    }
}

// ---------------------------------------------------------------------------
// Phase 2: proc_time sum / sumsq / nonzero-count via V_WMMA_F32_16X16X4_F32.
// A = ones(16x4)  =>  D[m][n] = colsum(B)[n] + C[m][n]  -> 16-bin tree reduction
// of 64 f32 per WMMA, accumulated through C. Wave handles a contiguous 2048-f32
// slice; lane loads 8B -> 256B per wave load (coalesced).
// ---------------------------------------------------------------------------
__global__ void pt_stats_kernel(const float* __restrict__ pt,
                                float* __restrict__ ws) {
    const int b      = blockIdx.x / PT_BLOCKS_PER_B;
    const int blkInB = blockIdx.x % PT_BLOCKS_PER_B;
    const int wave   = threadIdx.x >> 5;
    const int lane   = threadIdx.x & 31;
    const int waveIdx = blkInB * PT_WAVES_PER_BLK + wave;   // 0..127

    const float* base = pt + (size_t)b * PT_PER_B
                           + (size_t)waveIdx * (PT_CHUNKS_PER_WAVE * 64);

    const v2f ones = {1.0f, 1.0f};
    v8f accS = {};   // sum bins
    v8f accQ = {};   // sumsq bins
    v8f accC = {};   // nonzero-count bins

    for (int c = 0; c < PT_CHUNKS_PER_WAVE; ++c) {
        const float* p = base + (size_t)c * 64 + (size_t)lane * 2;
        __builtin_prefetch(p + 8 * 64, 0, 0);   // global_prefetch_b8, speculative
        v2f x = *reinterpret_cast<const v2f*>(p);
        v2f xx = x * x;
        v2f nz;
        nz[0] = (x[0] != 0.0f) ? 1.0f : 0.0f;
        nz[1] = (x[1] != 0.0f) ? 1.0f : 0.0f;
        // D = ones(16x4) * B(4x16) + C  : accumulates column sums of B into C
        accS = __builtin_amdgcn_wmma_f32_16x16x4_f32(false, ones, false, x,
                                                     (short)0, accS, false, false);
        accQ = __builtin_amdgcn_wmma_f32_16x16x4_f32(false, ones, false, xx,
                                                     (short)0, accQ, false, false);
        accC = __builtin_amdgcn_wmma_f32_16x16x4_f32(false, ones, false, nz,
                                                     (short)0, accC, false, false);
    }

    // VGPR0 of D holds colsum[lane%16] (rows identical since A==ones).
    // Sum over all 32 lanes counts every column twice -> scale by 0.5 (exact).
    float s  = accS[0];
    float q  = accQ[0];
    float cn = accC[0];
    for (int off = 16; off >= 1; off >>= 1) {
        s  += __shfl_down(s,  off);
        q  += __shfl_down(q,  off);
        cn += __shfl_down(cn, off);
    }
    if (lane == 0) {
        float* o = ws + WS_PT_PART + (size_t)(b * PT_WAVES_PER_B + waveIdx) * 3;
        o[0] = s  * 0.5f;
        o[1] = q  * 0.5f;
        o[2] = cn * 0.5f;
    }
}

// ---------------------------------------------------------------------------
// Phase 2b: deterministically combine the 128 wave partials per batch.
// ---------------------------------------------------------------------------
__global__ void pt_reduce_kernel(float* __restrict__ ws) {
    const int b = blockIdx.x;
    const int t = threadIdx.x;   // 0..127
    __shared__ float s[128][3];
    const float* p = ws + WS_PT_PART + (size_t)(b * PT_WAVES_PER_B + t) * 3;
    s[t][0] = p[0]; s[t][1] = p[1]; s[t][2] = p[2];
    __syncthreads();
    for (int off = 64; off >= 1; off >>= 1) {
        if (t < off) {
            s[t][0] += s[t + off][0];
            s[t][1] += s[t + off][1];
            s[t][2] += s[t + off][2];
        }
        __syncthreads();
    }
    if (t == 0) {
        float* o = ws + WS_PT_FINAL + b * 3;
        o[0] = s[0][0]; o[1] = s[0][1]; o[2] = s[0][2];
    }
}

// ---------------------------------------------------------------------------
// Phase 3: machine features (tiny): one 64-thread block per batch, end-to-end.
// ---------------------------------------------------------------------------
__global__ void mas_norm_kernel(const float* __restrict__ mas,
                                float* __restrict__ out) {
    const int b = blockIdx.x;
    const int m = threadIdx.x;   // 0..63
    __shared__ float ss[64][F_MA];
    __shared__ float sq[64][F_MA];

    float x[F_MA];
    const float* p = mas + (size_t)b * (NUM_MAS * F_MA) + (size_t)m * F_MA;
    for (int f = 0; f < F_MA; ++f) {
        x[f] = p[f];
        ss[m][f] = x[f];
        sq[m][f] = x[f] * x[f];
    }
    __syncthreads();
    for (int off = 32; off >= 1; off >>= 1) {
        if (m < off) {
            for (int f = 0; f < F_MA; ++f) {
                ss[m][f] += ss[m + off][f];
                sq[m][f] += sq[m + off][f];
            }
        }
        __syncthreads();
    }
    float* o = out + (size_t)b * (NUM_MAS * F_MA) + (size_t)m * F_MA;
    for (int f = 0; f < F_MA; ++f) {
        float mean = ss[0][f] * (1.0f / NUM_MAS);
        float var  = (sq[0][f] - (float)NUM_MAS * mean * mean) * (1.0f / (NUM_MAS - 1));
        float sd   = sqrtf(fmaxf(var, 0.0f));
        o[f] = (x[f] - mean) / (sd + EPSF);
    }
}

// ---------------------------------------------------------------------------
// Phase 4: normalize raw_opes (float4 vectorized; F_OPE=8 -> f0 in {0,4}).
// ---------------------------------------------------------------------------
__global__ void opes_norm_kernel(const float* __restrict__ opes,
                                 const int* __restrict__ nums,
                                 const float* __restrict__ ws,
                                 float* __restrict__ out) {
    const size_t i4   = (size_t)blockIdx.x * blockDim.x + threadIdx.x;  // exact grid
    const size_t flat = i4 * 4;
    const int b  = (int)(flat >> 15);        // / (4096*8)
    const int f0 = (int)(flat & 7);          // 0 or 4
    const float n = (float)nums[b];

    const float4 v = reinterpret_cast<const float4*>(opes)[i4];
    const float vv[4] = {v.x, v.y, v.z, v.w};
    float o[4];
#pragma unroll
    for (int k = 0; k < 4; ++k) {
        const int f = f0 + k;
        float mean = ws[WS_OPES_SUM + b * 8 + f] / n;
        float var  = (ws[WS_OPES_SQ + b * 8 + f] - n * mean * mean) / (n - 1.0f);
        float sd   = sqrtf(fmaxf(var, 0.0f));
        o[k] = (vv[k] - mean) / (sd + EPSF);
    }
    float4 r; r.x = o[0]; r.y = o[1]; r.z = o[2]; r.w = o[3];
    reinterpret_cast<float4*>(out)[i4] = r;
}

// ---------------------------------------------------------------------------
// Phase 5: normalize proc_time nonzeros (float4 vectorized, 67M elements).
// ---------------------------------------------------------------------------
__global__ void pt_norm_kernel(const float* __restrict__ pt,
                               const float* __restrict__ ws,
                               float* __restrict__ out) {
    const size_t i4   = (size_t)blockIdx.x * blockDim.x + threadIdx.x;  // exact grid
    const size_t flat = i4 * 4;
    const int b = (int)(flat >> 18);         // / 262144
    const float s  = ws[WS_PT_FINAL + b * 3 + 0];
    const float q  = ws[WS_PT_FINAL + b * 3 + 1];
    const float cn = ws[WS_PT_FINAL + b * 3 + 2];
    const float m   = s / cn;
    const float var = (q - cn * m * m) / (cn - 1.0f);
    const float sd  = sqrtf(fmaxf(var, 0.0f));
    const float inv = 1.0f / (sd + EPSF);

    const float4 v = reinterpret_cast<const float4*>(pt)[i4];
    float4 r;
    r.x = (v.x != 0.0f) ? (v.x - m) * inv : 0.0f;
    r.y = (v.y != 0.0f) ? (v.y - m) * inv : 0.0f;
    r.z = (v.z != 0.0f) ? (v.z - m) * inv : 0.0f;
    r.w = (v.w != 0.0f) ? (v.w - m) * inv : 0.0f;
    reinterpret_cast<float4*>(out)[i4] = r;
}

// ---------------------------------------------------------------------------
extern "C" void kernel_launch(void* const* d_in, const int* in_sizes, int n_in,
                              void* d_out, int out_size, void* d_ws, size_t ws_size,
                              hipStream_t stream) {
    (void)in_sizes; (void)n_in; (void)out_size; (void)ws_size;

    const float* raw_opes = (const float*)d_in[0];
    const float* raw_mas  = (const float*)d_in[1];
    const float* proc     = (const float*)d_in[2];
    const int*   nums     = (const int*)d_in[3];
    float* out = (float*)d_out;
    float* ws  = (float*)d_ws;

    // stats passes
    opes_stats_kernel<<<BATCH, 256, 0, stream>>>(raw_opes, nums, ws);
    pt_stats_kernel<<<BATCH * PT_BLOCKS_PER_B, 256, 0, stream>>>(proc, ws);
    pt_reduce_kernel<<<BATCH, 128, 0, stream>>>(ws);

    // normalize passes
    mas_norm_kernel<<<BATCH, 64, 0, stream>>>(raw_mas, out + OUT_OFF_MAS);
    opes_norm_kernel<<<OPES_ELEMS / 4 / 256, 256, 0, stream>>>(raw_opes, nums, ws, out);
    pt_norm_kernel<<<(BATCH * PT_PER_B) / 4 / 256, 256, 0, stream>>>(proc, ws,
                                                                     out + OUT_OFF_PT);
}